// LinearMolaLayer_3977139716769
// MI455X (gfx1250) — compile-verified
//
#include <hip/hip_runtime.h>

// MoLA (top-2 mixture of LoRA experts), fp32, MI455X / gfx1250.
// x:[8192,2048] gate_w:[8,2048] lora_A:[8,16,2048] lora_B:[8,2048,16]
// out:[8192,2048].  Memory-bound (~130MB HBM traffic); compute via
// V_WMMA_F32_16X16X4_F32.

typedef __attribute__((ext_vector_type(2))) float v2f;
typedef __attribute__((ext_vector_type(8))) float v8f;

#define T_TOKENS 8192
#define D_DIM    2048
#define E_EXP    8
#define R_RANK   16
#define O_DIM    2048
#define SCALING  2.0f   // 32.0 / 16.0
#define DCHUNK   128

// ---------------------------------------------------------------------------
// Kernel 1: gating. One wave per token. combine[t][e] = routed weight *
// SCALING for the top-2 experts, 0 otherwise. Softmax denominator cancels in
// the top-2 renormalization, so we renormalize the raw exp() values directly.
// ---------------------------------------------------------------------------
__global__ __launch_bounds__(256) void gate_kernel(const float* __restrict__ x,
                                                   const float* __restrict__ gate_w,
                                                   float* __restrict__ combine) {
  const int wave = threadIdx.x >> 5;   // 8 waves per block -> 8 tokens
  const int lane = threadIdx.x & 31;
  const int t = blockIdx.x * 8 + wave;

  float acc[E_EXP];
#pragma unroll
  for (int e = 0; e < E_EXP; ++e) acc[e] = 0.f;

  const float* xrow = x + (size_t)t * D_DIM;
  for (int k = lane; k < D_DIM; k += 32) {
    const float xv = xrow[k];
#pragma unroll
    for (int e = 0; e < E_EXP; ++e)
      acc[e] += xv * gate_w[e * D_DIM + k];
  }

  // wave32 butterfly reduction for each of the 8 logits
#pragma unroll
  for (int e = 0; e < E_EXP; ++e) {
    float v = acc[e];
#pragma unroll
    for (int off = 16; off > 0; off >>= 1) v += __shfl_xor(v, off, 32);
    acc[e] = v;
  }

  if (lane == 0) {
    float mx = acc[0];
#pragma unroll
    for (int e = 1; e < E_EXP; ++e) mx = fmaxf(mx, acc[e]);
    float p[E_EXP];
#pragma unroll
    for (int e = 0; e < E_EXP; ++e) p[e] = __expf(acc[e] - mx);

    int e1 = 0;
#pragma unroll
    for (int e = 1; e < E_EXP; ++e)
      if (p[e] > p[e1]) e1 = e;  // strict > keeps first index on ties (top_k order)
    int e2 = (e1 == 0) ? 1 : 0;
#pragma unroll
    for (int e = 0; e < E_EXP; ++e)
      if (e != e1 && p[e] > p[e2]) e2 = e;

    const float denom = p[e1] + p[e2];
    const float w1 = (p[e1] / denom) * SCALING;
    const float w2 = (p[e2] / denom) * SCALING;
#pragma unroll
    for (int e = 0; e < E_EXP; ++e) {
      float c = 0.f;
      if (e == e1) c = w1;
      if (e == e2) c = w2;
      combine[(size_t)t * E_EXP + e] = c;
    }
  }
}

// ---------------------------------------------------------------------------
// Kernel 2: fused LoRA path. Block = 16-token tile, 256 threads = 8 waves.
// Stage 1: wave w computes h_w[16x16] = x_tile @ A_w^T via WMMA f32 16x16x4
//          (x staged in LDS; A_w is L2-resident).
// Scale rows of h_w by combine (includes SCALING), stash in LDS.
// Stage 2: wave w produces out columns [w*256, w*256+256): for each 16-wide
//          column tile, accumulate 8 experts x (K=16 as 4 WMMAs of K=4).
// ---------------------------------------------------------------------------
__global__ __launch_bounds__(256) void mola_kernel(const float* __restrict__ x,
                                                   const float* __restrict__ loraA,
                                                   const float* __restrict__ loraB,
                                                   const float* __restrict__ combine,
                                                   float* __restrict__ out) {
  __shared__ __align__(16) float xs[16][DCHUNK];          // 8 KB
  __shared__ __align__(16) float Hs[E_EXP][16][R_RANK];   // 8 KB

  const int tid  = threadIdx.x;
  const int wave = tid >> 5;       // expert id in stage 1, column slice in stage 2
  const int lane = tid & 31;
  const int m    = lane & 15;      // lane's M (stage1 A) / N (stage1 B) index
  const int half = lane >> 4;      // K-half selector for A/B fragments
  const size_t tbase = (size_t)blockIdx.x * 16;

  // ---- Stage 1: h_w = x_tile @ A_w^T --------------------------------------
  v8f h = {};
  const float* arow = loraA + ((size_t)wave * R_RANK + m) * D_DIM + 2 * half;

  for (int chunk = 0; chunk < D_DIM / DCHUNK; ++chunk) {
    const int dbase = chunk * DCHUNK;

    // cooperative coalesced load: 16 rows x 128 cols = 512 float4, 2 per thread
#pragma unroll
    for (int i = 0; i < 2; ++i) {
      const int idx = tid + 256 * i;   // 0..511
      const int row = idx >> 5;        // 0..15
      const int c4  = idx & 31;        // float4 within row
      const float4* src =
          reinterpret_cast<const float4*>(x + (tbase + row) * (size_t)D_DIM + dbase) + c4;
      reinterpret_cast<float4*>(&xs[row][0])[c4] = *src;
    }
    __syncthreads();

#pragma unroll 8
    for (int kc = 0; kc < DCHUNK / 4; ++kc) {
      const int k0 = kc * 4;
      // A frag (16x4): lane m = token, K = k0 + 2*half + {0,1}
      v2f a = *reinterpret_cast<const v2f*>(&xs[m][k0 + 2 * half]);
      // B frag (4x16): lane m = rank, K = k0 + 2*half + {0,1}
      v2f b = *reinterpret_cast<const v2f*>(arow + dbase + k0);
      h = __builtin_amdgcn_wmma_f32_16x16x4_f32(false, a, false, b, (short)0, h,
                                                false, false);
    }
    __syncthreads();
  }

  // Scale rows of h by routed weight (0 for unselected experts) and stash.
  // C/D layout: VGPR j, lane -> (M = j + 8*half, N = m).
#pragma unroll
  for (int j = 0; j < 8; ++j) {
    const float cw = combine[(tbase + j + 8 * half) * E_EXP + wave];
    Hs[wave][j + 8 * half][m] = h[j] * cw;
  }
  __syncthreads();

  // ---- Stage 2: out_tile = sum_e H_e @ B_e^T ------------------------------
  const int colbase = wave * (O_DIM / E_EXP);   // 256-wide slice per wave
  for (int nt = 0; nt < (O_DIM / E_EXP) / 16; ++nt) {
    const int o = colbase + nt * 16 + m;        // this lane's output column (N)
    v8f acc = {};
#pragma unroll
    for (int e = 0; e < E_EXP; ++e) {
      const float* brow = loraB + ((size_t)e * O_DIM + o) * R_RANK + 2 * half;
#pragma unroll
      for (int kc = 0; kc < R_RANK / 4; ++kc) {
        const int k0 = kc * 4;
        // A frag: M = token (lane m), K = rank = k0 + 2*half + {0,1}
        v2f a = *reinterpret_cast<const v2f*>(&Hs[e][m][k0 + 2 * half]);
        // B frag: N = output col (lane m), K = rank
        v2f b = *reinterpret_cast<const v2f*>(brow + k0);
        acc = __builtin_amdgcn_wmma_f32_16x16x4_f32(false, a, false, b, (short)0,
                                                    acc, false, false);
      }
    }
#pragma unroll
    for (int j = 0; j < 8; ++j) {
      out[(tbase + j + 8 * half) * (size_t)O_DIM + o] = acc[j];
    }
  }
}

// ---------------------------------------------------------------------------
extern "C" void kernel_launch(void* const* d_in, const int* in_sizes, int n_in,
                              void* d_out, int out_size, void* d_ws, size_t ws_size,
                              hipStream_t stream) {
  const float* x      = (const float*)d_in[0];
  const float* gate_w = (const float*)d_in[1];
  const float* loraA  = (const float*)d_in[2];
  const float* loraB  = (const float*)d_in[3];
  float* out     = (float*)d_out;
  float* combine = (float*)d_ws;   // 8192 * 8 * 4B = 256 KB scratch

  gate_kernel<<<T_TOKENS / 8, 256, 0, stream>>>(x, gate_w, combine);
  mola_kernel<<<T_TOKENS / 16, 256, 0, stream>>>(x, loraA, loraB, combine, out);
}